// GraphSAGE_23476291240662
// MI455X (gfx1250) — compile-verified
//
#include <hip/hip_runtime.h>
#include <hip/hip_bf16.h>

#define N_NODES  50000
#define N_EDGES  800000
#define N_GRAPHS 64
#define D_IN     90
#define D_PAD    96                 // 6 tiles of 16; 24 k-steps of 4
#define M_TILES  (N_NODES / 16)     // 3125, exact

typedef __attribute__((ext_vector_type(2))) float v2f;
typedef __attribute__((ext_vector_type(8))) float v8f;

// ---------------------------------------------------------------- utilities

__global__ void fill_zero(float* __restrict__ p, int n) {
    int i = blockIdx.x * blockDim.x + threadIdx.x;
    if (i < n) p[i] = 0.0f;
}

// xp[n][k] = x[n][k] (k<90) else 0
__global__ void pad_x(const float* __restrict__ x, float* __restrict__ xp) {
    int i = blockIdx.x * blockDim.x + threadIdx.x;
    if (i >= N_NODES * D_PAD) return;
    int n = i / D_PAD, k = i % D_PAD;
    xp[i] = (k < D_IN) ? x[n * D_IN + k] : 0.0f;
}

// dst[k][n] = W[n][k] for n,k < 90 else 0   (dst is 96x96, W is 90x90 row-major [out][in])
__global__ void transpose_pad_w(const float* __restrict__ W, float* __restrict__ dst) {
    int i = blockIdx.x * blockDim.x + threadIdx.x;
    if (i >= D_PAD * D_PAD) return;
    int k = i / D_PAD, n = i % D_PAD;
    dst[i] = (k < D_IN && n < D_IN) ? W[n * D_IN + k] : 0.0f;
}

__global__ void pad_bias(const float* __restrict__ b, float* __restrict__ dst) {
    int n = threadIdx.x;
    if (n < D_PAD) dst[n] = (n < D_IN) ? b[n] : 0.0f;
}

// ------------------------------------------------------------ edge scatter
// agg[dst] += x[src]; deg[dst] += 1.  All targets are L2-resident (~19 MB).

__global__ void scatter_edges(const int* __restrict__ src, const int* __restrict__ dst,
                              const float* __restrict__ xin,
                              float* __restrict__ agg, float* __restrict__ deg) {
    int e = blockIdx.x * blockDim.x + threadIdx.x;
    if (e >= N_EDGES) return;
    int s = src[e], d = dst[e];
    atomicAdd(&deg[d], 1.0f);
    const float4* xr = (const float4*)(xin + (size_t)s * D_PAD);
    float* ar = agg + (size_t)d * D_PAD;
#pragma unroll
    for (int q = 0; q < D_PAD / 4; ++q) {
        float4 v = xr[q];
        atomicAdd(ar + 4 * q + 0, v.x);
        atomicAdd(ar + 4 * q + 1, v.y);
        atomicAdd(ar + 4 * q + 2, v.z);
        atomicAdd(ar + 4 * q + 3, v.w);
    }
}

// ------------------------------------------------- fused SAGE layer (WMMA)
// out = relu( (agg/max(deg,1)) @ Wl^T + bl + x @ Wr^T )
// One wave per 16-row tile; V_WMMA_F32_16X16X4_F32, K swept 96/4 = 24 steps,
// two WMMAs per step (neighbor path + root path) into one f32 accumulator.

__global__ __launch_bounds__(32)
void sage_wmma(const float* __restrict__ agg, const float* __restrict__ deg,
               const float* __restrict__ xin,
               const float* __restrict__ WtL, const float* __restrict__ WtR,
               const float* __restrict__ bias, float* __restrict__ out) {
    __shared__ float sA[16 * D_PAD];   // deg-normalized neighbor aggregate tile
    __shared__ float sX[16 * D_PAD];   // root-feature tile

    const int m0   = blockIdx.x * 16;
    const int lane = threadIdx.x;          // 0..31, full wave => EXEC all ones
    const int half = lane >> 4;            // 0 | 1
    const int l16  = lane & 15;

    // Stage 16x96 A-tiles into LDS (coalesced; deg normalization fused here).
    for (int i = lane; i < 16 * D_PAD; i += 32) {
        int r = i / D_PAD;
        float inv = 1.0f / fmaxf(deg[m0 + r], 1.0f);
        sA[i] = agg[(size_t)(m0 + r) * D_PAD + (i % D_PAD)] * inv;
        sX[i] = xin[(size_t)(m0 + r) * D_PAD + (i % D_PAD)];
    }
    __syncthreads();

    for (int nt = 0; nt < 6; ++nt) {
        const int n0 = nt * 16;
        const float bn = bias[n0 + l16];   // column bias; same for all 8 acc rows
        v8f acc;
#pragma unroll
        for (int i = 0; i < 8; ++i) acc[i] = bn;

        for (int k0 = 0; k0 < D_PAD; k0 += 4) {
            const int ka = k0 + 2 * half;  // lanes 0-15: K=k0,k0+1 ; lanes 16-31: K=k0+2,k0+3
            v2f a, xr, bL, bR;
            a[0]  = sA[l16 * D_PAD + ka];       a[1]  = sA[l16 * D_PAD + ka + 1];
            xr[0] = sX[l16 * D_PAD + ka];       xr[1] = sX[l16 * D_PAD + ka + 1];
            bL[0] = WtL[ka * D_PAD + n0 + l16]; bL[1] = WtL[(ka + 1) * D_PAD + n0 + l16];
            bR[0] = WtR[ka * D_PAD + n0 + l16]; bR[1] = WtR[(ka + 1) * D_PAD + n0 + l16];
            acc = __builtin_amdgcn_wmma_f32_16x16x4_f32(false, a,  false, bL, (short)0, acc, false, false);
            acc = __builtin_amdgcn_wmma_f32_16x16x4_f32(false, xr, false, bR, (short)0, acc, false, false);
        }

        // ReLU + store: row = m0 + i + 8*half, col = n0 + l16
#pragma unroll
        for (int i = 0; i < 8; ++i) {
            int m = m0 + i + 8 * half;
            out[(size_t)m * D_PAD + n0 + l16] = fmaxf(acc[i], 0.0f);
        }
    }
}

// ----------------------------------------------------------------- pooling

__global__ void pool_accum(const float* __restrict__ h, const int* __restrict__ batch,
                           float* __restrict__ pooled, float* __restrict__ cnts) {
    int i = blockIdx.x * blockDim.x + threadIdx.x;
    if (i >= N_NODES * D_PAD) return;
    int n = i / D_PAD, k = i % D_PAD;
    int g = batch[n];
    atomicAdd(&pooled[g * D_PAD + k], h[i]);
    if (k == 0) atomicAdd(&cnts[g], 1.0f);
}

__global__ void final_mlp(const float* __restrict__ pooled, const float* __restrict__ cnts,
                          const float* __restrict__ Wf1, const float* __restrict__ bf1,
                          const float* __restrict__ Wf2, const float* __restrict__ bf2,
                          float* __restrict__ out) {
    int g = threadIdx.x;
    if (g >= N_GRAPHS) return;
    float inv = 1.0f / fmaxf(cnts[g], 1.0f);
    float acc = bf2[0];
    for (int j = 0; j < 32; ++j) {
        float s = bf1[j];
        for (int d = 0; d < D_IN; ++d)
            s += (pooled[g * D_PAD + d] * inv) * Wf1[j * D_IN + d];
        acc += fmaxf(s, 0.0f) * Wf2[j];
    }
    out[g] = acc;
}

// ------------------------------------------------------------------ driver

extern "C" void kernel_launch(void* const* d_in, const int* in_sizes, int n_in,
                              void* d_out, int out_size, void* d_ws, size_t ws_size,
                              hipStream_t stream) {
    (void)in_sizes; (void)n_in; (void)out_size; (void)ws_size;

    const float* x     = (const float*)d_in[0];
    const int*   ei    = (const int*)d_in[1];       // [2, E]
    const int*   batch = (const int*)d_in[2];
    const float* Wl[3] = {(const float*)d_in[3], (const float*)d_in[6], (const float*)d_in[9]};
    const float* bb[3] = {(const float*)d_in[4], (const float*)d_in[7], (const float*)d_in[10]};
    const float* Wr[3] = {(const float*)d_in[5], (const float*)d_in[8], (const float*)d_in[11]};
    const float* Wf1 = (const float*)d_in[12];
    const float* bf1 = (const float*)d_in[13];
    const float* Wf2 = (const float*)d_in[14];
    const float* bf2 = (const float*)d_in[15];
    float* out = (float*)d_out;

    const int* src = ei;
    const int* dst = ei + N_EDGES;

    // ---- workspace carve-up (256B aligned)
    char* ws = (char*)d_ws;
    size_t off = 0;
    auto take = [&](size_t bytes) {
        char* p = ws + off;
        off += (bytes + 255) & ~(size_t)255;
        return p;
    };
    const size_t featBytes = (size_t)N_NODES * D_PAD * sizeof(float);
    float* bufA  = (float*)take(featBytes);                 // current features (padded)
    float* bufB  = (float*)take(featBytes);                 // next features
    float* agg   = (float*)take(featBytes);                 // neighbor sums
    float* deg   = (float*)take((size_t)N_NODES * sizeof(float));
    float* WtLp[3], *WtRp[3], *bp[3];
    for (int l = 0; l < 3; ++l) {
        WtLp[l] = (float*)take((size_t)D_PAD * D_PAD * sizeof(float));
        WtRp[l] = (float*)take((size_t)D_PAD * D_PAD * sizeof(float));
        bp[l]   = (float*)take((size_t)D_PAD * sizeof(float));
    }
    float* pooled = (float*)take((size_t)N_GRAPHS * D_PAD * sizeof(float));
    float* cnts   = (float*)take((size_t)N_GRAPHS * sizeof(float));

    const int TPB = 256;
    const int featGrid = (N_NODES * D_PAD + TPB - 1) / TPB;
    const int wGrid    = (D_PAD * D_PAD + TPB - 1) / TPB;

    // ---- preprocess: pad features, transpose+pad weights, pad biases
    pad_x<<<featGrid, TPB, 0, stream>>>(x, bufA);
    for (int l = 0; l < 3; ++l) {
        transpose_pad_w<<<wGrid, TPB, 0, stream>>>(Wl[l], WtLp[l]);
        transpose_pad_w<<<wGrid, TPB, 0, stream>>>(Wr[l], WtRp[l]);
        pad_bias<<<1, D_PAD, 0, stream>>>(bb[l], bp[l]);
    }

    // ---- three SAGE layers
    float* cur = bufA;
    float* nxt = bufB;
    for (int l = 0; l < 3; ++l) {
        fill_zero<<<featGrid, TPB, 0, stream>>>(agg, N_NODES * D_PAD);
        fill_zero<<<(N_NODES + TPB - 1) / TPB, TPB, 0, stream>>>(deg, N_NODES);
        scatter_edges<<<(N_EDGES + TPB - 1) / TPB, TPB, 0, stream>>>(src, dst, cur, agg, deg);
        sage_wmma<<<M_TILES, 32, 0, stream>>>(agg, deg, cur, WtLp[l], WtRp[l], bp[l], nxt);
        float* t = cur; cur = nxt; nxt = t;
    }

    // ---- global mean pool + MLP head
    fill_zero<<<(N_GRAPHS * D_PAD + TPB - 1) / TPB, TPB, 0, stream>>>(pooled, N_GRAPHS * D_PAD);
    fill_zero<<<1, N_GRAPHS, 0, stream>>>(cnts, N_GRAPHS);
    pool_accum<<<featGrid, TPB, 0, stream>>>(cur, batch, pooled, cnts);
    final_mlp<<<1, N_GRAPHS, 0, stream>>>(pooled, cnts, Wf1, bf1, Wf2, bf2, out);
}